// SingleStreamMemoryBank_79224966742291
// MI455X (gfx1250) — compile-verified
//
#include <hip/hip_runtime.h>
#include <math.h>

// Problem constants (fixed by setup_inputs in the reference)
constexpr int Bn = 64;
constexpr int Kn = 4096;
constexpr int Dn = 512;
constexpr int KCH = 8;            // K chunks for retrieved partial sums
constexpr int KS  = Kn / KCH;     // 512 k's per chunk
constexpr int NCHUNK = Dn / 4;    // 128 WMMA K-chunks along D
constexpr int BPANEL_PER_B = NCHUNK * 32 * 2;   // floats per batch in B-panel

typedef float v2f __attribute__((ext_vector_type(2)));
typedef float v4f __attribute__((ext_vector_type(4)));
typedef float v8f __attribute__((ext_vector_type(8)));

// ---------------------------------------------------------------------------
// Kernel 0: L2-normalize query and item rows.  grid = 2*Bn blocks x 256 thr.
// ---------------------------------------------------------------------------
__global__ void normalize_rows_kernel(const float* __restrict__ q,
                                      const float* __restrict__ it,
                                      float* __restrict__ q_n,
                                      float* __restrict__ i_n) {
  const int row = blockIdx.x;            // 0..127
  const int t   = threadIdx.x;           // 0..255
  const float* src;
  float* dst;
  if (row < Bn) { src = q  + (size_t)row * Dn;        dst = q_n + (size_t)row * Dn; }
  else          { src = it + (size_t)(row - Bn) * Dn; dst = i_n + (size_t)(row - Bn) * Dn; }
  float a = src[t];
  float b = src[t + 256];
  __shared__ float red[256];
  red[t] = a * a + b * b;
  __syncthreads();
  for (int s = 128; s > 0; s >>= 1) {
    if (t < s) red[t] += red[t + s];
    __syncthreads();
  }
  float inv = 1.0f / fmaxf(sqrtf(red[0]), 1e-12f);
  dst[t]       = a * inv;
  dst[t + 256] = b * inv;
}

// ---------------------------------------------------------------------------
// Kernel 0b: build packed per-lane WMMA B-panel.
// Entry (b, chunk, lane) holds the v2f this lane feeds as B-fragment for
// D-chunk c = 4*chunk:  N=0 column -> q_n, N=1 column -> i_n, rest 0.
// Layout matches the documented 4x16 B striping (lanes 0-15: K=c,c+1;
// lanes 16-31: K=c+2,c+3).  grid = Bn blocks x 256 threads.
// ---------------------------------------------------------------------------
__global__ void build_bpanel_kernel(const float* __restrict__ q_n,
                                    const float* __restrict__ i_n,
                                    float* __restrict__ bpanel) {
  const int b = blockIdx.x;
  const int t = threadIdx.x;
  for (int e = t; e < NCHUNK * 32; e += 256) {
    const int chunk = e >> 5;
    const int lane  = e & 31;
    const int n     = lane & 15;
    const int half  = lane >> 4;
    const int c     = chunk * 4 + 2 * half;
    v2f v; v.x = 0.0f; v.y = 0.0f;
    if (n == 0)      { v.x = q_n[(size_t)b * Dn + c]; v.y = q_n[(size_t)b * Dn + c + 1]; }
    else if (n == 1) { v.x = i_n[(size_t)b * Dn + c]; v.y = i_n[(size_t)b * Dn + c + 1]; }
    *(v2f*)(bpanel + (size_t)b * BPANEL_PER_B + (size_t)e * 2) = v;
  }
}

// ---------------------------------------------------------------------------
// Kernel 1: fused WMMA similarity + bank copy.
// Each wave handles 16 bank rows of one batch.  Per D-chunk: one b64 bank
// load (A fragment), one coalesced b64 B-panel load, fused copy store,
// per-lane |m|^2 accumulation, and one v_wmma_f32_16x16x4_f32.
// grid = (Bn*Kn/16)/8 = 2048 blocks x 256 threads (8 waves/block).
// ---------------------------------------------------------------------------
__global__ void sim_copy_wmma_kernel(const float* __restrict__ bank,
                                     const float* __restrict__ bpanel,
                                     float* __restrict__ out_bank,
                                     float* __restrict__ sim_r,
                                     float* __restrict__ sim_u) {
  const int lane = threadIdx.x & 31;
  const int wid  = threadIdx.x >> 5;
  const int tile = blockIdx.x * 8 + wid;     // 0..16383, 256 tiles per batch
  const int b    = tile >> 8;
  const int k0   = (tile & 255) << 4;
  const int half = lane >> 4;                // 0: cols c,c+1   1: cols c+2,c+3
  const int n    = lane & 15;                // A row index / B column index

  const size_t rowBase = ((size_t)b * Kn + (size_t)(k0 + n)) * Dn + 2 * half;
  const float* rowPtr  = bank + rowBase;
  float*       outPtr  = out_bank + rowBase;
  const float* bpanRow = bpanel + (size_t)b * BPANEL_PER_B + (size_t)lane * 2;

  v8f acc = {};
  float nacc = 0.0f;

#pragma unroll 8
  for (int c = 0; c < Dn; c += 4) {
    v2f av = *(const v2f*)(rowPtr + c);
    *(v2f*)(outPtr + c) = av;                          // fused bank -> new_bank
    v2f bv = *(const v2f*)(bpanRow + (size_t)(c >> 2) * 64);
    nacc = fmaf(av.x, av.x, nacc);
    nacc = fmaf(av.y, av.y, nacc);
    acc = __builtin_amdgcn_wmma_f32_16x16x4_f32(
        false, av, false, bv, (short)0, acc, false, false);
  }

  // |m|^2 per row: lanes L and L+16 each hold half -> combine across halves.
  float n2full = nacc + __shfl_xor(nacc, 16, 32);   // lanes 0..15 hold rows 0..15

  // C layout: VGPR j, lanes 0-15 -> M=j, N=lane; lanes 16-31 -> M=j+8, N=lane-16.
  // dot_q rows 0..7 in lane 0, rows 8..15 in lane 16 (N==0); dot_i in lanes 1/17.
  for (int j = 0; j < 8; ++j) {
    const int row = j + 8 * half;
    float n2  = __shfl(n2full, row, 32);            // broadcast norm^2 of 'row'
    float inv = 1.0f / fmaxf(sqrtf(n2), 1e-12f);
    float val = acc[j] * inv;
    size_t o  = (size_t)b * Kn + (size_t)(k0 + row);
    if (n == 0)      sim_r[o] = val;
    else if (n == 1) sim_u[o] = val;
  }
}

// ---------------------------------------------------------------------------
// Kernel 2: per-batch softmax stats over sim_r + argmax over sim_u.
// grid = Bn blocks x 256 threads.
// ---------------------------------------------------------------------------
__global__ void stats_kernel(const float* __restrict__ sim_r,
                             const float* __restrict__ sim_u,
                             float* __restrict__ weights,
                             float* __restrict__ sumexp,
                             float* __restrict__ best_sim,
                             int* __restrict__ best_idx) {
  const int b = blockIdx.x;
  const int t = threadIdx.x;
  __shared__ float smax[256];
  __shared__ float sval[256];
  __shared__ int   sidx[256];

  float lmax = -INFINITY, lbest = -INFINITY;
  int lbidx = 0;
  for (int k = t; k < Kn; k += 256) {
    float r = sim_r[(size_t)b * Kn + k];
    lmax = fmaxf(lmax, r);
    float u = sim_u[(size_t)b * Kn + k];
    if (u > lbest) { lbest = u; lbidx = k; }
  }
  smax[t] = lmax; sval[t] = lbest; sidx[t] = lbidx;
  __syncthreads();
  for (int s = 128; s > 0; s >>= 1) {
    if (t < s) {
      smax[t] = fmaxf(smax[t], smax[t + s]);
      if (sval[t + s] > sval[t]) { sval[t] = sval[t + s]; sidx[t] = sidx[t + s]; }
    }
    __syncthreads();
  }
  const float m = smax[0];
  __syncthreads();

  float lsum = 0.0f;
  for (int k = t; k < Kn; k += 256) {
    float e = expf(sim_r[(size_t)b * Kn + k] - m);
    weights[(size_t)b * Kn + k] = e;               // unnormalized softmax
    lsum += e;
  }
  smax[t] = lsum;
  __syncthreads();
  for (int s = 128; s > 0; s >>= 1) {
    if (t < s) smax[t] += smax[t + s];
    __syncthreads();
  }
  if (t == 0) {
    sumexp[b]   = smax[0];
    best_sim[b] = sval[0];
    best_idx[b] = sidx[0];
  }
}

// ---------------------------------------------------------------------------
// Kernel 3: do_ema flag = (mean(best_sim) >= 0.8).  1 block x 64 threads.
// ---------------------------------------------------------------------------
__global__ void flag_kernel(const float* __restrict__ best_sim,
                            int* __restrict__ flag) {
  const int t = threadIdx.x;
  __shared__ float s[64];
  s[t] = best_sim[t];
  __syncthreads();
  for (int st = 32; st > 0; st >>= 1) {
    if (t < st) s[t] += s[t + st];
    __syncthreads();
  }
  if (t == 0) flag[0] = (s[0] * (1.0f / (float)Bn) >= 0.8f) ? 1 : 0;
}

// ---------------------------------------------------------------------------
// Kernel 4: retrieved partial sums.  grid = Bn*KCH blocks x 256 threads.
// 128-thread column group (float4 per thread) x 2-way k split per block.
// partial layout: [B][KCH*2][D]
// ---------------------------------------------------------------------------
__global__ void retrieved_partial_kernel(const float* __restrict__ bank,
                                         const float* __restrict__ weights,
                                         float* __restrict__ partial) {
  const int b    = blockIdx.x / KCH;
  const int ch   = blockIdx.x % KCH;
  const int t    = threadIdx.x;
  const int ksub = t >> 7;                 // 0 or 1
  const int d    = (t & 127) * 4;
  const float* bptr = bank + ((size_t)b * Kn + (size_t)ch * KS) * Dn + d;
  const float* wptr = weights + (size_t)b * Kn + (size_t)ch * KS;
  v4f acc = {};
  for (int kk = ksub; kk < KS; kk += 2) {
    float w  = wptr[kk];
    v4f   mv = *(const v4f*)(bptr + (size_t)kk * Dn);
    acc.x = fmaf(w, mv.x, acc.x);
    acc.y = fmaf(w, mv.y, acc.y);
    acc.z = fmaf(w, mv.z, acc.z);
    acc.w = fmaf(w, mv.w, acc.w);
  }
  *(v4f*)(partial + ((size_t)(b * KCH + ch) * 2 + ksub) * Dn + d) = acc;
}

// ---------------------------------------------------------------------------
// Kernel 5: reduce partials, divide by sumexp -> retrieved.  128 x 256.
// ---------------------------------------------------------------------------
__global__ void retrieved_final_kernel(const float* __restrict__ partial,
                                       const float* __restrict__ sumexp,
                                       float* __restrict__ retrieved) {
  const int idx = blockIdx.x * 256 + threadIdx.x;   // 0 .. Bn*Dn-1
  const int b = idx / Dn;
  const int d = idx % Dn;
  float s = 0.0f;
  for (int p = 0; p < KCH * 2; ++p)
    s += partial[((size_t)b * KCH * 2 + p) * Dn + d];
  retrieved[idx] = s / sumexp[b];
}

// ---------------------------------------------------------------------------
// Kernel 6: fix up one row per batch in new_bank.  grid = Bn blocks x 256.
// ---------------------------------------------------------------------------
__global__ void fixup_kernel(const float* __restrict__ bank,
                             const float* __restrict__ item,
                             const int* __restrict__ flag,
                             const int* __restrict__ best_idx,
                             const int* __restrict__ write_index,
                             float* __restrict__ out_bank) {
  const int b = blockIdx.x;
  const int t = threadIdx.x;
  const int doe = flag[0];
  const int row = doe ? best_idx[b] : write_index[0];
  const float* itm  = item + (size_t)b * Dn;
  const float* brow = bank + ((size_t)b * Kn + row) * Dn;
  float*       orow = out_bank + ((size_t)b * Kn + row) * Dn;
  for (int d = t; d < Dn; d += 256) {
    orow[d] = doe ? (0.5f * brow[d] + 0.5f * itm[d]) : itm[d];
  }
}

// ---------------------------------------------------------------------------
extern "C" void kernel_launch(void* const* d_in, const int* in_sizes, int n_in,
                              void* d_out, int out_size, void* d_ws, size_t ws_size,
                              hipStream_t stream) {
  (void)in_sizes; (void)n_in; (void)out_size; (void)ws_size;

  const float* query  = (const float*)d_in[0];     // [B, D]
  const float* item   = (const float*)d_in[1];     // [B, D]
  const float* bank   = (const float*)d_in[2];     // [B, K, D]
  const int*   widx   = (const int*)d_in[3];       // scalar

  float* retrieved = (float*)d_out;                // [B, D]
  float* new_bank  = retrieved + (size_t)Bn * Dn;  // [B, K, D]

  // Workspace carve-up (floats)
  float* fw       = (float*)d_ws;
  float* q_n      = fw;                                  // 32768
  float* i_n      = q_n + (size_t)Bn * Dn;               // 32768
  float* sim_r    = i_n + (size_t)Bn * Dn;               // 262144
  float* sim_u    = sim_r + (size_t)Bn * Kn;             // 262144
  float* weights  = sim_u + (size_t)Bn * Kn;             // 262144
  float* sumexp   = weights + (size_t)Bn * Kn;           // 64
  float* best_sim = sumexp + 64;                         // 64
  int*   best_idx = (int*)(best_sim + 64);               // 64
  int*   flag     = best_idx + 64;                       // 1 (padded to 64)
  float* bpanel   = (float*)(flag + 64);                 // Bn*128*32*2 = 524288
  float* partial  = bpanel + (size_t)Bn * BPANEL_PER_B;  // Bn*KCH*2*Dn = 524288

  // 0) normalize query + item
  normalize_rows_kernel<<<2 * Bn, 256, 0, stream>>>(query, item, q_n, i_n);

  // 0b) pack per-lane WMMA B fragments
  build_bpanel_kernel<<<Bn, 256, 0, stream>>>(q_n, i_n, bpanel);

  // 1) WMMA similarity (sim_r, sim_u) fused with bank -> new_bank copy
  sim_copy_wmma_kernel<<<(Bn * Kn / 16) / 8, 256, 0, stream>>>(
      bank, bpanel, new_bank, sim_r, sim_u);

  // 2) per-batch softmax stats + argmax
  stats_kernel<<<Bn, 256, 0, stream>>>(sim_r, sim_u, weights, sumexp,
                                       best_sim, best_idx);

  // 3) do_ema scalar flag
  flag_kernel<<<1, 64, 0, stream>>>(best_sim, flag);

  // 4) retrieved partial sums over K chunks
  retrieved_partial_kernel<<<Bn * KCH, 256, 0, stream>>>(bank, weights, partial);

  // 5) reduce partials -> retrieved
  retrieved_final_kernel<<<(Bn * Dn) / 256, 256, 0, stream>>>(partial, sumexp,
                                                              retrieved);

  // 6) fix up EMA / write_index row per batch in new_bank
  fixup_kernel<<<Bn, 256, 0, stream>>>(bank, item, flag, best_idx, widx,
                                       new_bank);
}